// CapLayer_71021579207395
// MI455X (gfx1250) — compile-verified
//
#include <hip/hip_runtime.h>
#include <math.h>

// CapsNet routing layer for MI455X (gfx1250), wave32.
//
// Algorithm: the 377 MB pred tensor is never materialized. Both routing
// einsums factor through the grouped 1x1 conv into per-group small GEMMs:
//   y_g[j,k]  = sum_p c_g[j,p]  * x_g[p,k]     (C[16x36] x Xt[36x16], K=36)
//   dB_g[p,j] = sum_k x_g[p,k]  * wv_g[k,j]    (Xt[48x8] x WV[8x16],  K=8)
// run on V_WMMA_F32_16X16X4_F32, one wave per group (32 waves/WG), one
// workgroup per batch element. All per-batch state (~310 KB) lives in the
// 320 KB WGP LDS; global traffic ~= read x once (18.9 MB) + write v.
//
// Codegen note: M/N padding lanes use index-CLAMPED unconditional LDS loads
// (duplicated real rows/cols). Since WMMA rows of A and cols of B only affect
// the same rows/cols of D (which we discard), no masking is needed, and the
// contraction dims are exact multiples of 4 — so no branches in WMMA loops.

typedef float v2f __attribute__((ext_vector_type(2)));
typedef float v8f __attribute__((ext_vector_type(8)));

#define G   32   // groups (NS)
#define P   36   // spatial positions (6x6)
#define K8  8    // in-channels per group
#define J   10   // out capsules
#define D   16   // out dim
#define ROUTE 3
#define NTHREADS 1024
#define CIN 256

__global__ __launch_bounds__(NTHREADS)
void caps_route_kernel(const float* __restrict__ x,    // [bs,256,6,6]
                       const float* __restrict__ Wc,   // [5120,8]
                       const float* __restrict__ bc,   // [5120]
                       float* __restrict__ out)        // [bs,10,16]
{
    __shared__ float xT[G * P * K8];        // Xt_g[p][k]              36 KB
    __shared__ float Wl[G * J * D * K8];    // W[g][j][d][k]          160 KB
    __shared__ float blog[G * P * J];       // routing logits b        45 KB
    __shared__ float cbuf[G * P * J];       // softmax c[g][p][j]      45 KB
    __shared__ float csum[G * J];           // sum_p c[g,p,j]
    __shared__ float ybuf[G * J * K8];      // y_g[j][k]               10 KB
    __shared__ float wvbuf[G * J * K8];     // wv_g[j][k]              10 KB
    __shared__ float bvbuf[G * J];          // bv_g[j]
    __shared__ float sv[J * D];             // s
    __shared__ float vbuf[J * D];           // v

    const int t    = threadIdx.x;
    const int b    = blockIdx.x;
    const int g    = t >> 5;        // wave id == group id for per-g steps
    const int lane = t & 31;
    const int m16  = lane & 15;     // WMMA row/col within half
    const int half = lane >> 4;     // A/B: selects K pair; C/D: +8 rows
    const int jc   = (m16 < J)  ? m16 : (J - 1);   // clamped capsule index
    const int kc   = (m16 < K8) ? m16 : (K8 - 1);  // clamped channel index

    // ---------------- load phase ----------------
    const float* xb = x + (size_t)b * (CIN * P);
    for (int idx = t; idx < CIN * P; idx += NTHREADS) {
        int ch = idx / P, p = idx % P;              // coalesced read
        xT[((ch >> 3) * P + p) * K8 + (ch & 7)] = xb[idx];
    }
    for (int idx = t; idx < G * J * D * K8; idx += NTHREADS) Wl[idx] = Wc[idx];
    for (int idx = t; idx < G * P * J;      idx += NTHREADS) blog[idx] = 0.0f;
    __syncthreads();

    for (int it = 0; it < ROUTE; ++it) {
        // ---- softmax over j per i=(g,p); store c into LDS ----
        for (int i = t; i < G * P; i += NTHREADS) {
            const float* bi = &blog[i * J];
            float mx = bi[0];
            #pragma unroll
            for (int j = 1; j < J; ++j) mx = fmaxf(mx, bi[j]);
            float den = 0.0f;
            float e[J];
            #pragma unroll
            for (int j = 0; j < J; ++j) { e[j] = __expf(bi[j] - mx); den += e[j]; }
            float rcp = 1.0f / den;
            #pragma unroll
            for (int j = 0; j < J; ++j) cbuf[i * J + j] = e[j] * rcp;
        }
        __syncthreads();

        // ---- csum[g][j] = sum_p c (bias contribution to s) ----
        for (int o = t; o < G * J; o += NTHREADS) {
            int gg = o / J, j = o % J;
            float acc = 0.0f;
            for (int p = 0; p < P; ++p) acc += cbuf[(gg * P + p) * J + j];
            csum[o] = acc;
        }
        __syncthreads();

        // ---- GEMM A (per wave g): Y[16x16] = C[16x36] * Xt[36x16], 9 k-steps
        {
            v8f acc = {};
            for (int ks = 0; ks < 9; ++ks) {
                int p0 = ks * 4 + half * 2;      // K = p0, p0+1 (K=36 exact)
                int i0 = g * P + p0;
                v2f a, bb;
                a.x  = cbuf[i0 * J + jc];        // A row m16 (clamped, dup rows >=J)
                a.y  = cbuf[(i0 + 1) * J + jc];
                bb.x = xT[i0 * K8 + kc];         // B col m16 (clamped, dup cols >=K8)
                bb.y = xT[(i0 + 1) * K8 + kc];
                acc = __builtin_amdgcn_wmma_f32_16x16x4_f32(
                          false, a, false, bb, (short)0, acc, false, false);
            }
            // D layout: lane holds rows r+8*half, col m16
            if (m16 < K8) {
                #pragma unroll
                for (int r = 0; r < 8; ++r) {
                    int j = r + 8 * half;
                    if (j < J) ybuf[(g * J + j) * K8 + m16] = acc[r];
                }
            }
        }
        __syncthreads();

        // ---- s[j,d] = sum_{g,k} W*y + sum_g csum*bias  (160 outputs) ----
        for (int o = t; o < J * D; o += NTHREADS) {
            int j = o / D, d = o % D;
            float acc = 0.0f;
            for (int gg = 0; gg < G; ++gg) {
                const float* wr = &Wl[((gg * J + j) * D + d) * K8];
                const float* yr = &ybuf[(gg * J + j) * K8];
                #pragma unroll
                for (int k = 0; k < K8; ++k) acc += wr[k] * yr[k];
                acc += csum[gg * J + j] * bc[(gg * J + j) * D + d];  // bias via L2
            }
            sv[o] = acc;
        }
        __syncthreads();

        // ---- squash: v = s * ||s|| / (1 + ||s||^2) ----
        for (int j = t; j < J; j += NTHREADS) {
            float n2 = 0.0f;
            #pragma unroll
            for (int d = 0; d < D; ++d) { float s = sv[j * D + d]; n2 += s * s; }
            float f = sqrtf(n2) / (1.0f + n2);
            #pragma unroll
            for (int d = 0; d < D; ++d) vbuf[j * D + d] = sv[j * D + d] * f;
        }
        __syncthreads();

        if (it == ROUTE - 1) break;   // last iteration's b-update is unused

        // ---- wv[g][j][k] = sum_d v*W ; bv[g][j] = sum_d v*bias (per wave g) ----
        for (int o = lane; o < J * K8; o += 32) {
            int j = o / K8, k = o % K8;
            float acc = 0.0f;
            #pragma unroll
            for (int d = 0; d < D; ++d)
                acc += vbuf[j * D + d] * Wl[((g * J + j) * D + d) * K8 + k];
            wvbuf[(g * J + j) * K8 + k] = acc;
        }
        if (lane < J) {
            float acc = 0.0f;
            #pragma unroll
            for (int d = 0; d < D; ++d)
                acc += vbuf[lane * D + d] * bc[(g * J + lane) * D + d];  // bias via L2
            bvbuf[g * J + lane] = acc;
        }
        __syncthreads();

        // ---- GEMM B (per wave g): dB[48x16] = Xt[48x8] * WV[8x16], update blog
        for (int mt = 0; mt < 3; ++mt) {
            v8f acc = {};
            int p  = mt * 16 + m16;
            int pe = (p < P) ? p : (P - 1);          // clamped row (dup rows >=P)
            for (int ks = 0; ks < 2; ++ks) {
                int k0 = ks * 4 + half * 2;          // K = k0, k0+1 (K=8 exact)
                v2f a, bb;
                a.x  = xT[(g * P + pe) * K8 + k0];
                a.y  = xT[(g * P + pe) * K8 + k0 + 1];
                bb.x = wvbuf[(g * J + jc) * K8 + k0];   // B col m16 (clamped)
                bb.y = wvbuf[(g * J + jc) * K8 + k0 + 1];
                acc = __builtin_amdgcn_wmma_f32_16x16x4_f32(
                          false, a, false, bb, (short)0, acc, false, false);
            }
            if (m16 < J) {
                float bv = bvbuf[g * J + m16];
                #pragma unroll
                for (int r = 0; r < 8; ++r) {
                    int pr = mt * 16 + r + 8 * half;
                    if (pr < P) blog[(g * P + pr) * J + m16] += acc[r] + bv;
                }
            }
        }
        __syncthreads();
    }

    // ---- write v (f32 [bs,10,16]) ----
    for (int o = t; o < J * D; o += NTHREADS)
        out[(size_t)b * (J * D) + o] = vbuf[o];
}

extern "C" void kernel_launch(void* const* d_in, const int* in_sizes, int n_in,
                              void* d_out, int out_size, void* d_ws, size_t ws_size,
                              hipStream_t stream) {
    const float* x  = (const float*)d_in[0];   // [bs,256,6,6] f32
    // d_in[1] = target (int32) — unused in forward
    const float* Wc = (const float*)d_in[2];   // [5120,8,1,1] f32
    const float* bc = (const float*)d_in[3];   // [5120] f32
    float* out = (float*)d_out;                // [bs,10,16] f32

    int bs = in_sizes[0] / (CIN * P);          // 512 for the reference shapes
    caps_route_kernel<<<dim3(bs), dim3(NTHREADS), 0, stream>>>(x, Wc, bc, out);
}